// HDR_Class_Capsule_35656818491740
// MI455X (gfx1250) — compile-verified
//
#include <hip/hip_runtime.h>
#include <math.h>

typedef __attribute__((ext_vector_type(2))) float v2f;
typedef __attribute__((ext_vector_type(8))) float v8f;

#define PCHUNK 64

// ---- WMMA helper: D = A(16x4 f32) * B(4x16 f32) + C(16x16 f32) -------------
__device__ __forceinline__ v8f wmma4(v2f a, v2f b, v8f c) {
    return __builtin_amdgcn_wmma_f32_16x16x4_f32(false, a, false, b, (short)0, c,
                                                 false, false);
}

// A-matrix 16x4 f32 layout: lanes 0-15 hold M=lane rows; VGPR0={K=0|K=2}, VGPR1={K=1|K=3}
// lane l: m = l&15, hi = l>>4  ->  elements (m, k0+2*hi) and (m, k0+2*hi+1)
__device__ __forceinline__ v2f loadA_W(const float* __restrict__ W, long pc,
                                       int k0, int m, int hi) {
    // W[p,c,d,i] row-major, Din=8
    return *(const v2f*)(W + (pc * 16 + m) * 8 + k0 + 2 * hi);
}
// B-matrix 4x16 f32 (K rows x N cols), N = b from X[b,p,i], Din=8
__device__ __forceinline__ v2f loadB_X(const float* __restrict__ X, int P, int p,
                                       int k0, int m, int hi) {
    return *(const v2f*)(X + ((long)m * P + p) * 8 + k0 + 2 * hi);
}

// ---- zero scratch ----------------------------------------------------------
__global__ void k_zero(float* __restrict__ p, int n) {
    int i = blockIdx.x * blockDim.x + threadIdx.x;
    if (i < n) p[i] = 0.0f;
}

// ---- hpred[b,q,c,d] = sum_e Wh[q,c,d,e] * prev[b,q,e]  (16x16x16 GEMM) -----
// stored in tile layout: hpred[(q*C+c)*256 + r*32 + lane]
__global__ __launch_bounds__(32) void k_hpred(const float* __restrict__ Wh,
                                              const float* __restrict__ prev,
                                              float* __restrict__ hpred,
                                              int C, int Q) {
    int c = blockIdx.x, q = blockIdx.y;
    int lane = threadIdx.x & 31, m = lane & 15, hi = lane >> 4;
    long qc = (long)q * C + c;
    v8f t = {0, 0, 0, 0, 0, 0, 0, 0};
#pragma unroll
    for (int k0 = 0; k0 < 16; k0 += 4) {
        v2f a = *(const v2f*)(Wh + (qc * 16 + m) * 16 + k0 + 2 * hi);        // Wh[q,c,d=m,e]
        v2f b = *(const v2f*)(prev + ((long)m * Q + q) * 16 + k0 + 2 * hi);  // prev[b=m,q,e]
        t = wmma4(a, b, t);
    }
#pragma unroll
    for (int r = 0; r < 8; ++r) hpred[qc * 256 + r * 32 + lane] = t[r];
}

// ---- pass 1: S[c][d][b] += sum_p pred (mixed), for uniform routing ---------
__global__ __launch_bounds__(32) void k_pred_sum(
    const float* __restrict__ W, const float* __restrict__ X,
    const float* __restrict__ hpred, const float* __restrict__ alpha,
    float* __restrict__ S, int C, int Q, int repQ, int P) {
    int c = blockIdx.x;
    int p0 = blockIdx.y * PCHUNK;
    int lane = threadIdx.x & 31, m = lane & 15, hi = lane >> 4;
    v8f acc = {0, 0, 0, 0, 0, 0, 0, 0};
    float hacc[8] = {0, 0, 0, 0, 0, 0, 0, 0};
    for (int p = p0; p < p0 + PCHUNK; ++p) {
        long pc = (long)p * C + c;
        v2f a0 = loadA_W(W, pc, 0, m, hi);
        v2f a1 = loadA_W(W, pc, 4, m, hi);
        v2f b0 = loadB_X(X, P, p, 0, m, hi);
        v2f b1 = loadB_X(X, P, p, 4, m, hi);
        acc = wmma4(a0, b0, acc);
        acc = wmma4(a1, b1, acc);
        if (Q > 0 && p < repQ) {
            const float* hp = hpred + ((long)(p % Q) * C + c) * 256 + lane;
#pragma unroll
            for (int r = 0; r < 8; ++r) hacc[r] += hp[r * 32];
        }
    }
    float av = 1.0f, bv = 0.0f;
    if (Q > 0) { av = 1.0f / (1.0f + __expf(-alpha[0])); bv = 1.0f - av; }
#pragma unroll
    for (int r = 0; r < 8; ++r) {
        int d = r + 8 * hi;
        atomicAdd(&S[c * 256 + d * 16 + m], av * acc[r] + bv * hacc[r]);
    }
}

// ---- squash: per (b,c) over d; optional tile-layout and [b][c][d] outputs --
__global__ void k_squash(const float* __restrict__ S, float scale,
                         float* __restrict__ out_tile,  // [c][d][b] or null
                         float* __restrict__ out_bcd,   // [b][c][d] or null
                         float* __restrict__ prev_bcd,  // [b][c][d] or null
                         int B, int C) {
    int t = blockIdx.x * blockDim.x + threadIdx.x;
    if (t >= B * C) return;
    int b = t / C, c = t % C;
    float v[16], sq = 0.0f;
#pragma unroll
    for (int d = 0; d < 16; ++d) {
        v[d] = S[c * 256 + d * 16 + b] * scale;
        sq += v[d] * v[d];
    }
    float coef = (sq / (1.0f + sq)) / sqrtf(sq + 1e-7f);
#pragma unroll
    for (int d = 0; d < 16; ++d) {
        float o = coef * v[d];
        if (out_tile) out_tile[c * 256 + d * 16 + b] = o;
        if (out_bcd) out_bcd[((long)b * C + c) * 16 + d] = o;
        if (prev_bcd) prev_bcd[((long)b * C + c) * 16 + d] = o;
    }
}

// ---- pass 2: raw[b,p,c] = sum_d pred(mixed) * out0[b,c,d] ------------------
__global__ __launch_bounds__(32) void k_raw(
    const float* __restrict__ W, const float* __restrict__ X,
    const float* __restrict__ hpred, const float* __restrict__ alpha,
    const float* __restrict__ out0, float* __restrict__ raw,
    int C, int Q, int repQ, int P) {
    int c = blockIdx.x;
    int p0 = blockIdx.y * PCHUNK;
    int lane = threadIdx.x & 31, m = lane & 15, hi = lane >> 4;
    float av = 1.0f, bv = 0.0f;
    if (Q > 0) { av = 1.0f / (1.0f + __expf(-alpha[0])); bv = 1.0f - av; }
    for (int p = p0; p < p0 + PCHUNK; ++p) {
        long pc = (long)p * C + c;
        v2f a0 = loadA_W(W, pc, 0, m, hi);
        v2f a1 = loadA_W(W, pc, 4, m, hi);
        v2f b0 = loadB_X(X, P, p, 0, m, hi);
        v2f b1 = loadB_X(X, P, p, 4, m, hi);
        v8f t = {0, 0, 0, 0, 0, 0, 0, 0};
        t = wmma4(a0, b0, t);
        t = wmma4(a1, b1, t);
        const float* hp = (Q > 0 && p < repQ)
                              ? hpred + ((long)(p % Q) * C + c) * 256 + lane
                              : nullptr;
        float s = 0.0f;
#pragma unroll
        for (int r = 0; r < 8; ++r) {
            float pv = av * t[r] + (hp ? bv * hp[r * 32] : 0.0f);
            s += pv * out0[c * 256 + (r + 8 * hi) * 16 + m];
        }
        s += __shfl_xor(s, 16);  // combine d-halves (lanes l and l^16 share b)
        if (lane < 16) raw[((long)m * P + p) * C + c] = s;
    }
}

// ---- softmax prep over c per (b,p): max and inverse denominator ------------
__global__ void k_smax(const float* __restrict__ raw, float* __restrict__ mbuf,
                       float* __restrict__ dinv, int C, long BP) {
    long t = (long)blockIdx.x * blockDim.x + threadIdx.x;
    if (t >= BP) return;
    const float* r = raw + t * C;
    float mx = r[0];
    for (int c = 1; c < C; ++c) mx = fmaxf(mx, r[c]);
    float s = 0.0f;
    for (int c = 0; c < C; ++c) s += __expf(r[c] - mx);
    mbuf[t] = mx;
    dinv[t] = 1.0f / s;
}

// ---- pass 3: S[c][d][b] += sum_p softmax_c(raw) * pred(mixed) --------------
__global__ __launch_bounds__(32) void k_ws1(
    const float* __restrict__ W, const float* __restrict__ X,
    const float* __restrict__ hpred, const float* __restrict__ alpha,
    const float* __restrict__ raw, const float* __restrict__ mbuf,
    const float* __restrict__ dinv, float* __restrict__ S,
    int C, int Q, int repQ, int P) {
    int c = blockIdx.x;
    int p0 = blockIdx.y * PCHUNK;
    int lane = threadIdx.x & 31, m = lane & 15, hi = lane >> 4;
    float av = 1.0f, bv = 0.0f;
    if (Q > 0) { av = 1.0f / (1.0f + __expf(-alpha[0])); bv = 1.0f - av; }
    float acc[8] = {0, 0, 0, 0, 0, 0, 0, 0};
    for (int p = p0; p < p0 + PCHUNK; ++p) {
        long pc = (long)p * C + c;
        v2f a0 = loadA_W(W, pc, 0, m, hi);
        v2f a1 = loadA_W(W, pc, 4, m, hi);
        v2f b0 = loadB_X(X, P, p, 0, m, hi);
        v2f b1 = loadB_X(X, P, p, 4, m, hi);
        v8f t = {0, 0, 0, 0, 0, 0, 0, 0};
        t = wmma4(a0, b0, t);
        t = wmma4(a1, b1, t);
        long bp = (long)m * P + p;  // b = m (both lane halves share b)
        float rw = __expf(raw[bp * C + c] - mbuf[bp]) * dinv[bp];
        const float* hp = (Q > 0 && p < repQ)
                              ? hpred + ((long)(p % Q) * C + c) * 256 + lane
                              : nullptr;
#pragma unroll
        for (int r = 0; r < 8; ++r) {
            float pv = av * t[r] + (hp ? bv * hp[r * 32] : 0.0f);
            acc[r] += rw * pv;
        }
    }
#pragma unroll
    for (int r = 0; r < 8; ++r)
        atomicAdd(&S[c * 256 + (r + 8 * hi) * 16 + m], acc[r]);
}

extern "C" void kernel_launch(void* const* d_in, const int* in_sizes, int n_in,
                              void* d_out, int out_size, void* d_ws, size_t ws_size,
                              hipStream_t stream) {
    const float* X = (const float*)d_in[0];
    const float* Wl[3] = {(const float*)d_in[1], (const float*)d_in[2],
                          (const float*)d_in[3]};
    const float* Whs[2] = {(const float*)d_in[4], (const float*)d_in[5]};
    const float* al[2] = {(const float*)d_in[6], (const float*)d_in[7]};
    const int B = 16, P = 2048, D = 16;
    const int nc[3] = {2, 20, 100};

    float* ws = (float*)d_ws;
    float* Sbuf = ws;    ws += 100 * 256;        // sum_p pred (tile layout)
    float* Wsbuf = ws;   ws += 100 * 256;        // iter-1 weighted sum
    float* out0b = ws;   ws += 100 * 256;        // squash(iter0) tile layout
    float* prevb = ws;   ws += 16 * 100 * 16;    // prev level output [b][q][e]
    float* hpredb = ws;  ws += 20 * 100 * 256;   // hpred tiles
    float* mbuf = ws;    ws += (long)B * P;
    float* dbuf = ws;    ws += (long)B * P;
    float* rawb = ws;    ws += (long)B * P * 100;

    float* outp = (float*)d_out;
    for (int h = 0; h < 3; ++h) {
        int C = nc[h];
        int Q = 0, repQ = 0;
        const float* alpha = nullptr;
        if (h > 0) {
            Q = nc[h - 1];
            repQ = (P / Q) * Q;
            alpha = al[h - 1];
            k_hpred<<<dim3(C, Q), 32, 0, stream>>>(Whs[h - 1], prevb, hpredb, C, Q);
        }
        k_zero<<<(C * 256 + 255) / 256, 256, 0, stream>>>(Sbuf, C * 256);
        k_pred_sum<<<dim3(C, P / PCHUNK), 32, 0, stream>>>(Wl[h], X, hpredb, alpha,
                                                           Sbuf, C, Q, repQ, P);
        k_squash<<<(B * C + 63) / 64, 64, 0, stream>>>(Sbuf, 1.0f / (float)C, out0b,
                                                       nullptr, nullptr, B, C);
        k_raw<<<dim3(C, P / PCHUNK), 32, 0, stream>>>(Wl[h], X, hpredb, alpha, out0b,
                                                      rawb, C, Q, repQ, P);
        k_smax<<<(B * P + 255) / 256, 256, 0, stream>>>(rawb, mbuf, dbuf, C,
                                                        (long)B * P);
        k_zero<<<(C * 256 + 255) / 256, 256, 0, stream>>>(Wsbuf, C * 256);
        k_ws1<<<dim3(C, P / PCHUNK), 32, 0, stream>>>(Wl[h], X, hpredb, alpha, rawb,
                                                      mbuf, dbuf, Wsbuf, C, Q, repQ, P);
        k_squash<<<(B * C + 63) / 64, 64, 0, stream>>>(Wsbuf, 1.0f, nullptr, outp,
                                                       prevb, B, C);
        outp += (long)B * C * D;
    }
}